// CombinedModel_77335181131998
// MI455X (gfx1250) — compile-verified
//
#include <hip/hip_runtime.h>
#include <hip/hip_bf16.h>
#include <math.h>
#include <cstdint>

typedef float v2f __attribute__((ext_vector_type(2)));
typedef float v8f __attribute__((ext_vector_type(8)));
typedef unsigned int u32x4 __attribute__((ext_vector_type(4)));
typedef int i32x4 __attribute__((ext_vector_type(4)));
typedef int i32x8 __attribute__((ext_vector_type(8)));

__device__ inline float fsigmoid(float x) { return 1.0f / (1.0f + __expf(-x)); }

// ---------------------------------------------------------------------------
// Tensor Data Mover (TDM) support: async DMA of a 2D tile from global to LDS,
// with LDS row padding, tracked by TENSORcnt. Falls back to manual staging if
// the builtin is unavailable.
// ---------------------------------------------------------------------------
#if defined(__HIP_DEVICE_COMPILE__) && __has_builtin(__builtin_amdgcn_tensor_load_to_lds)
#define HAVE_TDM 1
#else
#define HAVE_TDM 0
#endif

#if HAVE_TDM
// 2D tiled tensor load: tile_rows x tile_cols fp32 elements starting at gptr
// (row stride = row_stride elements) -> LDS at lds_off, padding pad_amount
// dwords after every pad_interval dwords (codes per D# spec). tensor_cols /
// tensor_rows give the remaining valid extent from the tile start so the TDM
// zero-fills out-of-bounds reads (K-tail handling).
__device__ __forceinline__ void tdm_load_tile_2d(
    const float* gptr, unsigned lds_off,
    unsigned tensor_cols, unsigned tensor_rows,
    unsigned tile_cols, unsigned tile_rows,
    unsigned row_stride,
    unsigned pad_interval_code, unsigned pad_amount_code) {
  u32x4 g0;
  i32x8 g1;
  i32x4 gz;
  unsigned long long ga = (unsigned long long)(uintptr_t)gptr;
  // D# group 0: count=1 | lds_addr | global_addr[56:0] | type=2 ("image")
  g0[0] = 1u;
  g0[1] = lds_off;
  g0[2] = (unsigned)ga;
  g0[3] = (unsigned)((ga >> 32) & 0x1FFFFFFull) | (2u << 30);
  // D# group 1: data_size=2 (4B), pad_enable, pad codes, dims, dim0 stride
  g1[0] = (int)((2u << 16) | (1u << 20) |
                (pad_interval_code << 22) | (pad_amount_code << 25));
  g1[1] = (int)((tensor_cols & 0xFFFFu) << 16);              // tensor_dim0 lo
  g1[2] = (int)((tensor_cols >> 16) | ((tensor_rows & 0xFFFFu) << 16));
  g1[3] = (int)((tensor_rows >> 16) | (tile_cols << 16));    // tile_dim0
  g1[4] = (int)(tile_rows & 0xFFFFu);                        // tile_dim1, dim2=0
  g1[5] = (int)row_stride;                                   // dim0 stride lo32
  g1[6] = 0;
  g1[7] = 0;
  gz[0] = 0; gz[1] = 0; gz[2] = 0; gz[3] = 0;
#if __clang_major__ >= 23
  i32x8 z8;
  z8[0]=0; z8[1]=0; z8[2]=0; z8[3]=0; z8[4]=0; z8[5]=0; z8[6]=0; z8[7]=0;
  __builtin_amdgcn_tensor_load_to_lds(g0, g1, gz, gz, z8, 0);
#else
  __builtin_amdgcn_tensor_load_to_lds(g0, g1, gz, gz, 0);
#endif
}
#endif

// ---------------------------------------------------------------------------
// WMMA fp32 GEMM:  C[M,N] = A[M,K] * B[K,N] + bias[N]
// Block tile 64x128, 256 threads = 8 waves (wave32). Wave (wm, wn):
//   wm = wave%4 -> 16-row tile, wn = wave/4 -> 64-col group (4 x 16-col tiles)
// V_WMMA_F32_16X16X4_F32 on LDS-staged tiles. With TDM: double-buffered async
// tile DMA (wave 0 drives TENSORcnt), one barrier per K-tile. LDS rows padded
// (A: 33, B: 132 dwords) against bank conflicts — TDM pad feature matches.
// Requires M%64==0, N%128==0 (true at all call sites); K arbitrary (zero tail).
// ---------------------------------------------------------------------------
template <int KC>
__global__ __launch_bounds__(256) void gemm_bias_f32(
    const float* __restrict__ A, const float* __restrict__ B,
    const float* __restrict__ bias, float* __restrict__ C,
    int M, int N, int K) {
  static_assert(KC == 32, "pad codes below assume KC==32");
#if HAVE_TDM
  __shared__ float As[2][64][KC + 1];
  __shared__ float Bs[2][KC][128 + 4];
#else
  __shared__ float As[1][64][KC + 1];
  __shared__ float Bs[1][KC][128 + 4];
#endif

  const int tid  = threadIdx.x;
  const int lane = tid & 31;
  const int wave = tid >> 5;
  const int wm   = wave & 3;
  const int wn   = wave >> 2;
  const int lr   = lane & 15;
  const int lh   = lane >> 4;  // 0/1: selects K pair within group-of-4
  const int m0   = blockIdx.y * 64;
  const int n0   = blockIdx.x * 128;

  v8f acc[4];
#pragma unroll
  for (int t = 0; t < 4; ++t)
#pragma unroll
    for (int v = 0; v < 8; ++v) acc[t][v] = 0.0f;

  const int ntile = (K + KC - 1) / KC;

#if HAVE_TDM
  if (wave == 0) {
    // prologue: DMA tile 0 into buffer 0
    tdm_load_tile_2d(A + (size_t)m0 * K, (unsigned)(uintptr_t)&As[0][0][0],
                     (unsigned)K, (unsigned)(M - m0), (unsigned)KC, 64u,
                     (unsigned)K, /*interval 32dw*/ 4u, /*pad 1dw*/ 0u);
    tdm_load_tile_2d(B + n0, (unsigned)(uintptr_t)&Bs[0][0][0],
                     (unsigned)(N - n0), (unsigned)K, 128u, (unsigned)KC,
                     (unsigned)N, /*interval 128dw*/ 6u, /*pad 4dw*/ 3u);
  }
#endif

  for (int kt = 0; kt < ntile; ++kt) {
#if HAVE_TDM
    const int cur = kt & 1;
    if (wave == 0) __builtin_amdgcn_s_wait_tensorcnt(0);  // tile kt landed
    __syncthreads();  // publish LDS to all waves; prior reads all drained
    if (wave == 0 && kt + 1 < ntile) {
      const int kn = (kt + 1) * KC;  // DMA tile kt+1 while computing kt
      tdm_load_tile_2d(A + (size_t)m0 * K + kn,
                       (unsigned)(uintptr_t)&As[cur ^ 1][0][0],
                       (unsigned)(K - kn), (unsigned)(M - m0), (unsigned)KC,
                       64u, (unsigned)K, 4u, 0u);
      tdm_load_tile_2d(B + (size_t)kn * N + n0,
                       (unsigned)(uintptr_t)&Bs[cur ^ 1][0][0],
                       (unsigned)(N - n0), (unsigned)(K - kn), 128u,
                       (unsigned)KC, (unsigned)N, 6u, 3u);
    }
#else
    const int cur = 0;
    const int k0 = kt * KC;
    for (int idx = tid; idx < 64 * KC; idx += 256) {
      int r = idx / KC, c = idx % KC;
      int gk = k0 + c;
      As[0][r][c] = (gk < K) ? A[(size_t)(m0 + r) * K + gk] : 0.0f;
    }
    for (int idx = tid; idx < KC * 128; idx += 256) {
      int r = idx / 128, c = idx % 128;
      int gk = k0 + r;
      Bs[0][r][c] = (gk < K) ? B[(size_t)gk * N + (n0 + c)] : 0.0f;
    }
    __syncthreads();
#endif

#pragma unroll
    for (int kg = 0; kg < KC / 4; ++kg) {
      const int kb = kg * 4 + 2 * lh;
      v2f a;
      a[0] = As[cur][wm * 16 + lr][kb];
      a[1] = As[cur][wm * 16 + lr][kb + 1];
#pragma unroll
      for (int t = 0; t < 4; ++t) {
        v2f b;
        b[0] = Bs[cur][kb][wn * 64 + t * 16 + lr];
        b[1] = Bs[cur][kb + 1][wn * 64 + t * 16 + lr];
        acc[t] = __builtin_amdgcn_wmma_f32_16x16x4_f32(
            false, a, false, b, (short)0, acc[t], false, false);
      }
    }
#if !HAVE_TDM
    __syncthreads();
#endif
  }

#pragma unroll
  for (int t = 0; t < 4; ++t) {
    const int col = n0 + wn * 64 + t * 16 + lr;
    const float bv = bias[col];
#pragma unroll
    for (int v = 0; v < 8; ++v) {
      const int row = m0 + wm * 16 + v + 8 * lh;
      C[(size_t)row * N + col] = acc[t][v] + bv;
    }
  }
}

// ---------------------------------------------------------------------------
// LSTM recurrence with WMMA. One block = 16 batch rows, all 4H gate columns.
// Per step t: z[16,4H] = xz[:,t,:] + h[16,H] @ Wh[H,4H]   (WMMA f32 16x16x4)
// then elementwise gate math; h kept in LDS, c in registers.
// ACT: 0 = relu, 1 = tanh. Gate order i,f,g,o (Keras).
// ---------------------------------------------------------------------------
template <int H, int ACT, bool WRITE_ALL, bool WRITE_LAST>
__global__ __launch_bounds__(256) void lstm_rec_kernel(
    const float* __restrict__ xz,   // [B*T, 4H] (row = b*T + t)
    const float* __restrict__ Wh,   // [H, 4H]
    float* __restrict__ all_h,      // [B*T, H] (row = b*T + t) or unused
    float* __restrict__ last_h,     // [B, H] or unused
    int T) {
  constexpr int G   = 4 * H;
  constexpr int NT  = G / 16;          // 16 (H=64) or 32 (H=128)
  constexpr int TPW = NT / 8;          // tiles per wave: 2 or 4
  constexpr int EPT = (16 * H) / 256;  // state elems per thread: 4 or 8

  __shared__ float h_sh[16 * H];
  __shared__ float z_sh[16 * G];

  const int tid  = threadIdx.x;
  const int lane = tid & 31;
  const int wave = tid >> 5;
  const int lr   = lane & 15;
  const int lh   = lane >> 4;
  const int b0   = blockIdx.x * 16;

  float c_reg[EPT];
#pragma unroll
  for (int i = 0; i < EPT; ++i) c_reg[i] = 0.0f;
  for (int i = tid; i < 16 * H; i += 256) h_sh[i] = 0.0f;
  __syncthreads();

  for (int t = 0; t < T; ++t) {
    v8f acc[TPW];
    // init accumulators with precomputed input projection xz_t
#pragma unroll
    for (int nt = 0; nt < TPW; ++nt) {
      const int col = (wave * TPW + nt) * 16 + lr;
#pragma unroll
      for (int v = 0; v < 8; ++v) {
        const int r = v + 8 * lh;
        acc[nt][v] = xz[((size_t)(b0 + r) * T + t) * G + col];
      }
    }
    // z += h @ Wh   (K = H in groups of 4); Wh stays L2/WGP$-resident
#pragma unroll
    for (int kg = 0; kg < H / 4; ++kg) {
      const int kb = kg * 4 + 2 * lh;
      v2f a;
      a[0] = h_sh[lr * H + kb];
      a[1] = h_sh[lr * H + kb + 1];
#pragma unroll
      for (int nt = 0; nt < TPW; ++nt) {
        const int col = (wave * TPW + nt) * 16 + lr;
        v2f b;
        b[0] = Wh[(size_t)kb * G + col];
        b[1] = Wh[(size_t)(kb + 1) * G + col];
        acc[nt] = __builtin_amdgcn_wmma_f32_16x16x4_f32(
            false, a, false, b, (short)0, acc[nt], false, false);
      }
    }
    // spill z to LDS so each thread can gather its i/f/g/o quadruple
#pragma unroll
    for (int nt = 0; nt < TPW; ++nt) {
      const int col = (wave * TPW + nt) * 16 + lr;
#pragma unroll
      for (int v = 0; v < 8; ++v) {
        const int r = v + 8 * lh;
        z_sh[r * G + col] = acc[nt][v];
      }
    }
    __syncthreads();

#pragma unroll
    for (int i = 0; i < EPT; ++i) {
      const int e = tid + i * 256;
      const int r = e / H, j = e % H;
      const float zi = z_sh[r * G + j];
      const float zf = z_sh[r * G + H + j];
      const float zg = z_sh[r * G + 2 * H + j];
      const float zo = z_sh[r * G + 3 * H + j];
      const float ig = fsigmoid(zi);
      const float fg = fsigmoid(zf);
      const float gg = (ACT == 0) ? fmaxf(zg, 0.0f) : tanhf(zg);
      const float og = fsigmoid(zo);
      const float cc = fg * c_reg[i] + ig * gg;
      c_reg[i] = cc;
      const float ca = (ACT == 0) ? fmaxf(cc, 0.0f) : tanhf(cc);
      const float hh = og * ca;
      h_sh[r * H + j] = hh;
      if constexpr (WRITE_ALL) {
        all_h[((size_t)(b0 + r) * T + t) * H + j] = hh;
      }
      if constexpr (WRITE_LAST) {
        if (t == T - 1) last_h[(size_t)(b0 + r) * H + j] = hh;
      }
    }
    __syncthreads();
  }
}

// ---------------------------------------------------------------------------
// GRU input projection: gxz[BT,24] = hs[BT,64] @ gWx[64,24] + gb[0]
// Tiny (50 MFLOP) -> plain VALU kernel.
// ---------------------------------------------------------------------------
__global__ void gru_pre_kernel(const float* __restrict__ hs,
                               const float* __restrict__ gWx,
                               const float* __restrict__ gb,
                               float* __restrict__ gxz, int BT) {
  const int idx = blockIdx.x * blockDim.x + threadIdx.x;
  if (idx >= BT * 24) return;
  const int row = idx / 24, col = idx % 24;
  float s = gb[col];
#pragma unroll
  for (int k = 0; k < 64; ++k) s += hs[(size_t)row * 64 + k] * gWx[k * 24 + col];
  gxz[idx] = s;
}

// GRU recurrence (reset_after=True), H=8, one thread per batch row.
__global__ __launch_bounds__(256) void gru_rec_kernel(
    const float* __restrict__ gxz, const float* __restrict__ gWh,
    const float* __restrict__ gb, float* __restrict__ g_last, int T) {
  const int b = blockIdx.x * blockDim.x + threadIdx.x;  // 0..255
  float h[8];
#pragma unroll
  for (int j = 0; j < 8; ++j) h[j] = 0.0f;
  for (int t = 0; t < T; ++t) {
    const float* xz = gxz + ((size_t)b * T + t) * 24;
    float rec[24];
#pragma unroll
    for (int c = 0; c < 24; ++c) {
      float s = gb[24 + c];
#pragma unroll
      for (int k = 0; k < 8; ++k) s += h[k] * gWh[k * 24 + c];
      rec[c] = s;
    }
    float hn[8];
#pragma unroll
    for (int j = 0; j < 8; ++j) {
      const float z  = fsigmoid(xz[j] + rec[j]);
      const float r  = fsigmoid(xz[8 + j] + rec[8 + j]);
      const float hh = tanhf(xz[16 + j] + r * rec[16 + j]);
      hn[j] = z * h[j] + (1.0f - z) * hh;
    }
#pragma unroll
    for (int j = 0; j < 8; ++j) h[j] = hn[j];
  }
#pragma unroll
  for (int j = 0; j < 8; ++j) g_last[b * 8 + j] = h[j];
}

// Dense heads + concat + final softmax. One thread per batch row (B=256).
__global__ __launch_bounds__(256) void heads_kernel(
    const float* __restrict__ h3, const float* __restrict__ g,
    const float* __restrict__ kD1w, const float* __restrict__ kD1b,
    const float* __restrict__ kD2w, const float* __restrict__ kD2b,
    const float* __restrict__ iDw, const float* __restrict__ iDb,
    const float* __restrict__ fW, const float* __restrict__ fb,
    float* __restrict__ out) {
  const int b = blockIdx.x * blockDim.x + threadIdx.x;
  float d1[64];
#pragma unroll 4
  for (int j = 0; j < 64; ++j) {
    float s = kD1b[j];
#pragma unroll 8
    for (int k = 0; k < 64; ++k) s += h3[b * 64 + k] * kD1w[k * 64 + j];
    d1[j] = fmaxf(s, 0.0f);
  }
  float key[32];
#pragma unroll 4
  for (int j = 0; j < 32; ++j) {
    float s = kD2b[j];
#pragma unroll 8
    for (int k = 0; k < 64; ++k) s += d1[k] * kD2w[k * 32 + j];
    key[j] = fmaxf(s, 0.0f);
  }
  float img[8];
#pragma unroll
  for (int j = 0; j < 8; ++j) {
    float s = iDb[j];
#pragma unroll
    for (int k = 0; k < 8; ++k) s += g[b * 8 + k] * iDw[k * 8 + j];
    img[j] = fmaxf(s, 0.0f);
  }
  float logit[10];
#pragma unroll
  for (int j = 0; j < 10; ++j) {
    float s = fb[j];
#pragma unroll
    for (int k = 0; k < 8; ++k) s += img[k] * fW[k * 10 + j];
#pragma unroll
    for (int k = 0; k < 32; ++k) s += key[k] * fW[(8 + k) * 10 + j];
    logit[j] = s;
  }
  float m = logit[0];
#pragma unroll
  for (int j = 1; j < 10; ++j) m = fmaxf(m, logit[j]);
  float e[10], sum = 0.0f;
#pragma unroll
  for (int j = 0; j < 10; ++j) { e[j] = __expf(logit[j] - m); sum += e[j]; }
  const float inv = 1.0f / sum;
#pragma unroll
  for (int j = 0; j < 10; ++j) out[b * 10 + j] = e[j] * inv;
}

// ---------------------------------------------------------------------------
extern "C" void kernel_launch(void* const* d_in, const int* in_sizes, int n_in,
                              void* d_out, int out_size, void* d_ws, size_t ws_size,
                              hipStream_t stream) {
  const float* keypoint = (const float*)d_in[0];
  const float* img      = (const float*)d_in[1];
  const float* kW1x = (const float*)d_in[2];
  const float* kW1h = (const float*)d_in[3];
  const float* kb1  = (const float*)d_in[4];
  const float* kW2x = (const float*)d_in[5];
  const float* kW2h = (const float*)d_in[6];
  const float* kb2  = (const float*)d_in[7];
  const float* kW3x = (const float*)d_in[8];
  const float* kW3h = (const float*)d_in[9];
  const float* kb3  = (const float*)d_in[10];
  const float* kD1w = (const float*)d_in[11];
  const float* kD1b = (const float*)d_in[12];
  const float* kD2w = (const float*)d_in[13];
  const float* kD2b = (const float*)d_in[14];
  const float* iWx  = (const float*)d_in[15];
  const float* iWh  = (const float*)d_in[16];
  const float* ib   = (const float*)d_in[17];
  const float* gWx  = (const float*)d_in[18];
  const float* gWh  = (const float*)d_in[19];
  const float* gb   = (const float*)d_in[20];
  const float* iDw  = (const float*)d_in[21];
  const float* iDb  = (const float*)d_in[22];
  const float* fW   = (const float*)d_in[23];
  const float* fb   = (const float*)d_in[24];

  const int B = 256, T = 64, BT = B * T;

  // workspace layout (sequentially reused):
  //   [0, 32MB)      xz ping buffer (max 16384 x 512 f32) / later GRU xz
  //   [32MB, 40MB)   h buffer (max 16384 x 128 f32)
  //   [40MB, +64KB)  h3_last (256 x 64)
  //   [+64KB, +8KB)  g_last (256 x 8)
  char* ws = (char*)d_ws;
  float* xz   = (float*)(ws);
  float* hbuf = (float*)(ws + ((size_t)32 << 20));
  float* h3   = (float*)(ws + ((size_t)40 << 20));
  float* g    = (float*)(ws + ((size_t)40 << 20) + ((size_t)64 << 10));
  float* gxz  = xz;  // xz region is free by the time the GRU projection runs

  // ---- keypoint branch: 3 relu LSTMs ----
  gemm_bias_f32<32><<<dim3(256 / 128, BT / 64), 256, 0, stream>>>(
      keypoint, kW1x, kb1, xz, BT, 256, 1662);
  lstm_rec_kernel<64, 0, true, false><<<B / 16, 256, 0, stream>>>(
      xz, kW1h, hbuf, nullptr, T);
  gemm_bias_f32<32><<<dim3(512 / 128, BT / 64), 256, 0, stream>>>(
      hbuf, kW2x, kb2, xz, BT, 512, 64);
  lstm_rec_kernel<128, 0, true, false><<<B / 16, 256, 0, stream>>>(
      xz, kW2h, hbuf, nullptr, T);
  gemm_bias_f32<32><<<dim3(256 / 128, BT / 64), 256, 0, stream>>>(
      hbuf, kW3x, kb3, xz, BT, 256, 128);
  lstm_rec_kernel<64, 0, false, true><<<B / 16, 256, 0, stream>>>(
      xz, kW3h, nullptr, h3, T);

  // ---- img branch: tanh LSTM -> GRU ----
  gemm_bias_f32<32><<<dim3(256 / 128, BT / 64), 256, 0, stream>>>(
      img, iWx, ib, xz, BT, 256, 2048);
  lstm_rec_kernel<64, 1, true, false><<<B / 16, 256, 0, stream>>>(
      xz, iWh, hbuf, nullptr, T);
  gru_pre_kernel<<<(BT * 24 + 255) / 256, 256, 0, stream>>>(hbuf, gWx, gb, gxz, BT);
  gru_rec_kernel<<<1, 256, 0, stream>>>(gxz, gWh, gb, g, T);

  // ---- heads + softmax ----
  heads_kernel<<<1, 256, 0, stream>>>(h3, g, kD1w, kD1b, kD2w, kD2b,
                                      iDw, iDb, fW, fb, (float*)d_out);
}